// SingleQubitReuploadingTorch_79594333929756
// MI455X (gfx1250) — compile-verified
//
#include <hip/hip_runtime.h>

#define TPB 256
#define MAX_LAYERS 64

typedef __attribute__((ext_vector_type(4))) float f4;
typedef int v4i __attribute__((vector_size(16)));

#define AS_GLOBAL __attribute__((address_space(1)))
#define AS_LDS    __attribute__((address_space(3)))

// Per-layer pre-folded parameters (32B for clean ds_load_b128 pairs):
//   hw0 = 0.5*w[l,0], hb0 = 0.5*b[l,0]   -> p  = phi/2   = fma(x0, hw0, hb0)
//   hw1 = 0.5*w[l,1], hb1 = 0.5*b[l,1]   -> ht = theta/2 = fma(x1, hw1, hb1)
//   co  = cos(omega/2), so = sin(omega/2),  omega = w[l,2] + b[l,2]
struct LayerParams { float hw0, hb0, hw1, hb1, co, so, pad0, pad1; };

__global__ __launch_bounds__(TPB)
void squbit_reupload_kernel(const float* __restrict__ X,
                            const float* __restrict__ W,
                            const float* __restrict__ Bb,
                            float* __restrict__ out,
                            int batch) {
    __shared__ float raw[2 * 3 * MAX_LAYERS];      // [0..191]=weights, [192..383]=biases
    __shared__ LayerParams P[MAX_LAYERS];

    const int tid = threadIdx.x;

    // ---- Stage raw weights+biases into LDS via CDNA5 async global->LDS DMA ----
#if __has_builtin(__builtin_amdgcn_global_load_async_to_lds_b128)
    if (tid < 96) {   // 96 lanes x 16B = 1536B = 384 floats
        const float* src = (tid < 48) ? (W + tid * 4) : (Bb + (tid - 48) * 4);
        float* dst = &raw[tid * 4];
        __builtin_amdgcn_global_load_async_to_lds_b128(
            (AS_GLOBAL v4i*)(float*)src,
            (AS_LDS    v4i*)dst,
            /*offset=*/0, /*cpol=*/0);
    }
  #if __has_builtin(__builtin_amdgcn_s_wait_asynccnt)
    __builtin_amdgcn_s_wait_asynccnt(0);
  #else
    asm volatile("s_wait_asynccnt 0x0" ::: "memory");
  #endif
#else
    if (tid < 192) {
        raw[tid]       = W[tid];
        raw[192 + tid] = Bb[tid];
    }
#endif
    __syncthreads();

    // ---- Fold per-layer constants (uniform work, 64 lanes) ----
    if (tid < MAX_LAYERS) {
        float w0 = raw[tid * 3 + 0], w1 = raw[tid * 3 + 1], w2 = raw[tid * 3 + 2];
        float b0 = raw[192 + tid * 3 + 0], b1 = raw[192 + tid * 3 + 1], b2 = raw[192 + tid * 3 + 2];
        float o  = 0.5f * (w2 + b2);                 // omega/2 (layer constant)
        LayerParams lp;
        lp.hw0 = 0.5f * w0;  lp.hb0 = 0.5f * b0;
        lp.hw1 = 0.5f * w1;  lp.hb1 = 0.5f * b1;
        lp.co  = __cosf(o);  lp.so  = __sinf(o);
        lp.pad0 = 0.0f;      lp.pad1 = 0.0f;
        P[tid] = lp;
    }
    __syncthreads();

    const int b = blockIdx.x * TPB + tid;
    if (b >= batch) return;

    const float x0 = X[2 * b + 0];
    const float x1 = X[2 * b + 1];

    // state = [1+0i, 0+0i]
    float s0r = 1.0f, s0i = 0.0f, s1r = 0.0f, s1i = 0.0f;

    f4* __restrict__ outv = (f4*)out;     // states[l][b] = {s0.re, s0.im, s1.re, s1.im}

#pragma unroll 4
    for (int l = 0; l < MAX_LAYERS; ++l) {
        const LayerParams p = P[l];                          // broadcast ds_load_b128 x2

        const float pp = fmaf(x0, p.hw0, p.hb0);             // phi/2
        const float ht = fmaf(x1, p.hw1, p.hb1);             // theta/2

        const float sp = __sinf(pp), cp = __cosf(pp);
        const float st = __sinf(ht), ct = __cosf(ht);

        // cos/sin of (phi +- omega)/2 via angle addition with layer constants
        const float cpco = cp * p.co, spso = sp * p.so;
        const float spco = sp * p.co, cpso = cp * p.so;
        const float cpo = cpco - spso, spo = spco + cpso;    // cos/sin(ppo)
        const float cmo = cpco + spso, smo = spco - cpso;    // cos/sin(pmo)

        // m00 = A - iB, m01 = -C - iD, m10 = C - iD, m11 = A + iB
        const float A = ct * cpo, Bv = ct * spo;
        const float C = st * cmo, D  = st * smo;

        const float n0r = A * s0r + Bv * s0i - C * s1r + D  * s1i;
        const float n0i = A * s0i - Bv * s0r - C * s1i - D  * s1r;
        const float n1r = C * s0r + D  * s0i + A * s1r - Bv * s1i;
        const float n1i = C * s0i - D  * s0r + A * s1i + Bv * s1r;
        s0r = n0r; s0i = n0i; s1r = n1r; s1i = n1i;

        f4 v; v.x = s0r; v.y = s0i; v.z = s1r; v.w = s1i;
        __builtin_nontemporal_store(v, &outv[(size_t)l * (size_t)batch + (size_t)b]);
    }
}

extern "C" void kernel_launch(void* const* d_in, const int* in_sizes, int n_in,
                              void* d_out, int out_size, void* d_ws, size_t ws_size,
                              hipStream_t stream) {
    const float* X  = (const float*)d_in[0];   // (B, 2) f32
    const float* W  = (const float*)d_in[1];   // (64, 3) f32
    const float* Bb = (const float*)d_in[2];   // (64, 3) f32
    float* out = (float*)d_out;                // (64, B, 2) complex64 -> f32 flat

    const int batch = in_sizes[0] / 2;
    const int blocks = (batch + TPB - 1) / TPB;
    squbit_reupload_kernel<<<blocks, TPB, 0, stream>>>(X, W, Bb, out, batch);
}